// OVR_CSP_53300544143821
// MI455X (gfx1250) — compile-verified
//
#include <hip/hip_runtime.h>

typedef __attribute__((ext_vector_type(2))) float v2f;
typedef __attribute__((ext_vector_type(8))) float v8f;

#define NN 64
#define BB 16
#define CC 22
#define TT 4000
#define MM 16
#define NWAVES 8
#define NTILES (TT / 16) /* 250 */

__global__ __launch_bounds__(NWAVES * 32)
void OVR_CSP_covpow_wmma_f32(const float* __restrict__ xfb,
                             const float* __restrict__ WT,
                             float* __restrict__ out)
{
    const int nb   = blockIdx.x;       // n * BB + b
    const int b    = nb & (BB - 1);
    const int tid  = threadIdx.x;
    const int lane = tid & 31;
    const int wave = tid >> 5;
    const int half = lane >> 4;        // 0: lanes 0-15, 1: lanes 16-31
    const int tcol = lane & 15;        // N (= t) column within the tile

    const float* __restrict__ X  = xfb + (size_t)nb * CC * TT;  // 22 x 4000, t fast
    const float* __restrict__ Wb = WT  + (size_t)b  * MM * CC;  // 16 x 22, c fast

    // ---- Preload A fragments: WT[b] as 6 chunks of 16x4 (C padded 22 -> 24) ----
    // 32-bit A 16x4 layout: lanes 0-15 hold (M=lane, K=k0,k0+1), lanes 16-31 K=k0+2,k0+3.
    // Branchless: clamp address to a valid element, select zero for padded columns.
    v2f afrag[6];
#pragma unroll
    for (int k = 0; k < 6; ++k) {
        const int c0 = 4 * k + 2 * half;
        const int c1 = c0 + 1;
        const float a0 = Wb[tcol * CC + (c0 < CC ? c0 : CC - 1)];
        const float a1 = Wb[tcol * CC + (c1 < CC ? c1 : CC - 1)];
        afrag[k].x = (c0 < CC) ? a0 : 0.0f;   // zeros at c=22,23 make the padded
        afrag[k].y = (c1 < CC) ? a1 : 0.0f;   // B rows irrelevant (A-side padding)
    }

    v8f sums = {};  // per-lane running sum of Z^2 over this wave's t-columns

    // ---- Stream T in 16-column tiles; never materialize Z ----
#pragma unroll 2
    for (int tile = wave; tile < NTILES; tile += NWAVES) {
        // Base for chunks 0..4: includes the per-lane half-row offset.
        const float* __restrict__ XtA = X + tile * 16 + tcol + (size_t)(2 * half) * TT;
        // Base for the K-residual chunk 5: rows clamped to 20/21 for ALL lanes.
        // half=1 lanes would need rows 22,23, but afrag[5]==0 there, so any
        // in-bounds data is fine -> fully branchless, single load clause.
        const float* __restrict__ XtB = X + tile * 16 + tcol + (size_t)20 * TT;

        v8f acc = {};
#pragma unroll
        for (int k = 0; k < 5; ++k) {
            v2f bfrag;
            bfrag.x = __builtin_nontemporal_load(&XtA[(size_t)(4 * k) * TT]);
            bfrag.y = __builtin_nontemporal_load(&XtA[(size_t)(4 * k + 1) * TT]);
            acc = __builtin_amdgcn_wmma_f32_16x16x4_f32(
                false, afrag[k], false, bfrag, (short)0, acc, false, false);
        }
        {
            v2f bfrag;
            bfrag.x = __builtin_nontemporal_load(&XtB[0]);
            bfrag.y = __builtin_nontemporal_load(&XtB[(size_t)TT]);
            acc = __builtin_amdgcn_wmma_f32_16x16x4_f32(
                false, afrag[5], false, bfrag, (short)0, acc, false, false);
        }
        // Z tile (16x16 f32): VGPR r -> M = r + 8*half, lane -> t. Square in-register.
#pragma unroll
        for (int r = 0; r < 8; ++r)
            sums[r] += acc[r] * acc[r];
    }

    // ---- Reduce over t within each 16-lane half (xor masks stay inside halves) ----
#pragma unroll
    for (int off = 8; off >= 1; off >>= 1) {
#pragma unroll
        for (int r = 0; r < 8; ++r)
            sums[r] += __shfl_xor(sums[r], off, 32);
    }

    __shared__ float red[NWAVES][MM];
    if (tcol == 0) {
#pragma unroll
        for (int r = 0; r < 8; ++r)
            red[wave][r + 8 * half] = sums[r];   // lane0 -> m=0..7, lane16 -> m=8..15
    }
    __syncthreads();

    // ---- Final: combine waves, normalize over m, log ----
    if (wave == 0) {
        const int m = lane & 15;
        float d = 0.0f;
#pragma unroll
        for (int w = 0; w < NWAVES; ++w) d += red[w][m];
        float total = d;
#pragma unroll
        for (int off = 8; off >= 1; off >>= 1)
            total += __shfl_xor(total, off, 32);   // sum over all 16 m (per half)
        if (lane < 16)
            out[(size_t)nb * MM + m] = __logf(d / total);
    }
}

extern "C" void kernel_launch(void* const* d_in, const int* in_sizes, int n_in,
                              void* d_out, int out_size, void* d_ws, size_t ws_size,
                              hipStream_t stream)
{
    (void)in_sizes; (void)n_in; (void)out_size; (void)d_ws; (void)ws_size;
    const float* xfb = (const float*)d_in[0];
    const float* WT  = (const float*)d_in[1];
    float* out       = (float*)d_out;

    OVR_CSP_covpow_wmma_f32<<<NN * BB, NWAVES * 32, 0, stream>>>(xfb, WT, out);
}